// MambaSSM_7851200217480
// MI455X (gfx1250) — compile-verified
//
#include <hip/hip_runtime.h>
#include <hip/hip_bf16.h>

// y[i,j] = U[i] * d[j], where U = inclusive prefix sum of u (len S=1024)
// and d[j] = sum_n (u[j]*W_B[n]+b_B[n]) * (u[j]*W_C[n]+b_C[n]) over N=64.
// (A / delta terms cancel exactly along the cumsum axis — see analysis.)

#define S_LEN 1024
#define N_DIM 64

typedef __attribute__((ext_vector_type(2))) float v2f;
typedef __attribute__((ext_vector_type(8))) float v8f;

// ---------------------------------------------------------------------------
// Kernel 1: one 1024-thread block.
//   - LDS Hillis-Steele inclusive scan of u -> U
//   - per-j quadratic dot over N=64 -> d[j]
// ---------------------------------------------------------------------------
__global__ void __launch_bounds__(S_LEN)
mamba_prep_kernel(const float* __restrict__ u,
                  const float* __restrict__ W_B, const float* __restrict__ b_B,
                  const float* __restrict__ W_C, const float* __restrict__ b_C,
                  float* __restrict__ U, float* __restrict__ dvec) {
    __shared__ float s0[S_LEN];
    __shared__ float s1[S_LEN];
    const int t = threadIdx.x;

    const float uv = u[t];
    s0[t] = uv;
    __syncthreads();

    float* src = s0;
    float* dst = s1;
    #pragma unroll
    for (int off = 1; off < S_LEN; off <<= 1) {
        float v = src[t];
        if (t >= off) v += src[t - off];
        dst[t] = v;
        __syncthreads();
        float* tmp = src; src = dst; dst = tmp;
    }
    U[t] = src[t];

    // d[j] = sum_n B[j,n] * C[j,n]   (B = u*W_B + b_B, C = u*W_C + b_C)
    float acc = 0.0f;
    #pragma unroll 8
    for (int n = 0; n < N_DIM; ++n) {
        const float Bjn = fmaf(uv, W_B[n], b_B[n]);
        const float Cjn = fmaf(uv, W_C[n], b_C[n]);
        acc = fmaf(Bjn, Cjn, acc);
    }
    dvec[t] = acc;
}

// ---------------------------------------------------------------------------
// Kernel 2: rank-1 outer product via V_WMMA_F32_16X16X4_F32.
// One wave32 per 16x16 output tile; grid = (S/16, S/16).
// A(16x4): U[m] replicated into all 4 K slots (both VGPRs, both lane halves).
// B(4x16): d[n]/4 replicated likewise (divide by 4 is exact in fp).
// => D[m,n] = sum_k U[m]*d[n]/4 = U[m]*d[n], independent of the K-slot map.
// EXEC stays all-ones (WMMA requirement).
// ---------------------------------------------------------------------------
__global__ void __launch_bounds__(32)
mamba_outer_wmma_kernel(const float* __restrict__ U,
                        const float* __restrict__ dvec,
                        float* __restrict__ y) {
    const int lane = threadIdx.x;          // 0..31, wave32
    const int j0   = blockIdx.x * 16;      // output column tile
    const int i0   = blockIdx.y * 16;      // output row tile
    const int m    = lane & 15;            // row (A) / col (B) index in tile

    const float ua = U[i0 + m];
    const float db = 0.25f * dvec[j0 + m];

    v2f a; a.x = ua; a.y = ua;             // all K slots = U[m]
    v2f b; b.x = db; b.y = db;             // all K slots = d[n]/4
    v8f c = {};

    // D = A(16x4) x B(4x16) + 0  ->  v_wmma_f32_16x16x4_f32
    c = __builtin_amdgcn_wmma_f32_16x16x4_f32(
            /*neg_a=*/false, a, /*neg_b=*/false, b,
            /*c_mod=*/(short)0, c, /*reuse_a=*/false, /*reuse_b=*/false);

    // C/D layout: VGPR r -> rows (i0+r) [lanes 0-15] and (i0+r+8) [lanes 16-31],
    // column = j0 + (lane&15). Half-wave rows are 64B-contiguous stores.
    const int  rbase = i0 + ((lane >> 4) << 3);
    float* out = y + (size_t)rbase * S_LEN + (j0 + m);
    #pragma unroll
    for (int r = 0; r < 8; ++r)
        out[(size_t)r * S_LEN] = c[r];
}

// ---------------------------------------------------------------------------
extern "C" void kernel_launch(void* const* d_in, const int* in_sizes, int n_in,
                              void* d_out, int out_size, void* d_ws, size_t ws_size,
                              hipStream_t stream) {
    // setup_inputs order: u, A, W_B, b_B, W_C, b_C, W_delta, b_delta
    const float* u   = (const float*)d_in[0];
    // d_in[1] (A), d_in[6] (W_delta), d_in[7] (b_delta): cancel exactly, unused.
    const float* W_B = (const float*)d_in[2];
    const float* b_B = (const float*)d_in[3];
    const float* W_C = (const float*)d_in[4];
    const float* b_C = (const float*)d_in[5];

    float* Uv = (float*)d_ws;        // S floats
    float* dv = Uv + S_LEN;          // S floats

    mamba_prep_kernel<<<1, S_LEN, 0, stream>>>(u, W_B, b_B, W_C, b_C, Uv, dv);

    dim3 grid(S_LEN / 16, S_LEN / 16);   // 64 x 64 tiles
    mamba_outer_wmma_kernel<<<grid, 32, 0, stream>>>(Uv, dv, (float*)d_out);
}